// RelativeTemporalAttention_77876347011121
// MI455X (gfx1250) — compile-verified
//
#include <hip/hip_runtime.h>
#include <math.h>

namespace {
constexpr int kD = 256;          // embed dim
constexpr int kS = 2048;         // sequence length
constexpr int kW = 5;            // temporal window
constexpr int kRows = 16;        // rows (b,s positions) per block / per wave
constexpr int kRowStride = kD + 8;   // 264 floats: keeps 16B alignment, spreads LDS banks
constexpr float kEps = 1e-5f;

typedef __attribute__((ext_vector_type(2))) float v2f;
typedef __attribute__((ext_vector_type(8))) float v8f;
}

__device__ __forceinline__ float4 ld4(const float* p) { return *(const float4*)p; }
__device__ __forceinline__ void   st4(float* p, float4 v) { *(float4*)p = v; }
__device__ __forceinline__ float4 add4(float4 a, float4 b) {
  return make_float4(a.x + b.x, a.y + b.y, a.z + b.z, a.w + b.w);
}
__device__ __forceinline__ float4 sub4(float4 a, float4 b) {
  return make_float4(a.x - b.x, a.y - b.y, a.z - b.z, a.w - b.w);
}
// e + s1*u + s2*v   (fused)
__device__ __forceinline__ float4 biasfuse(float4 e, float s1, float4 u, float s2, float4 v) {
  return make_float4(fmaf(s1, u.x, fmaf(s2, v.x, e.x)),
                     fmaf(s1, u.y, fmaf(s2, v.y, e.y)),
                     fmaf(s1, u.z, fmaf(s2, v.z, e.z)),
                     fmaf(s1, u.w, fmaf(s2, v.w, e.w)));
}
// (x-mean)*rstd*g + b
__device__ __forceinline__ float4 lnorm4(float4 x, float mean, float rstd, float4 g, float4 b) {
  return make_float4(fmaf((x.x - mean) * rstd, g.x, b.x),
                     fmaf((x.y - mean) * rstd, g.y, b.y),
                     fmaf((x.z - mean) * rstd, g.z, b.z),
                     fmaf((x.w - mean) * rstd, g.w, b.w));
}

__global__ __launch_bounds__(32)
void rta_fused_kernel(const float* __restrict__ emb,
                      const int*   __restrict__ resp,
                      const float* __restrict__ relpos,   // [2W+1][D]
                      const float* __restrict__ resp_w,   // [D] (weight[:,0])
                      const float* __restrict__ gamma_,
                      const float* __restrict__ beta_,
                      float* __restrict__ out) {
  __shared__ __align__(16) float tile[kRows * kRowStride];
  __shared__ float stats[kRows * 2];   // (mean, rstd) per row
  __shared__ float respf[kRows];       // responses/4 per row

  const int  lane    = threadIdx.x;          // 0..31 (wave32)
  const int  col0    = lane * 8;             // this lane's 8 columns for compute phases
  const long rowbase = (long)blockIdx.x * kRows;   // flat (b*S + s) row index
  const int  s0      = (int)(rowbase % kS);  // kRows divides kS -> whole tile in one batch row

  // ---- Issue async copies of the whole 16x256 embedding tile into LDS.
  //      GLOBAL_LOAD_ASYNC_TO_LDS_B128, tracked by ASYNCcnt: 2 instructions per
  //      row, each a fully-coalesced 512B slab (lane l moves bytes [l*16,l*16+16)).
  //      Overlaps with the constant/bias preloads below. ----
  {
#pragma unroll
    for (int m = 0; m < kRows; ++m) {
#pragma unroll
      for (int c = 0; c < 2; ++c) {
        const float* gp = emb + (size_t)(rowbase + m) * kD + c * 128 + lane * 4;
        float*       lp = tile + m * kRowStride + c * 128 + lane * 4;
        unsigned           ldsa = (unsigned)(uintptr_t)lp;        // low 32b = LDS offset
        unsigned long long ga   = (unsigned long long)(uintptr_t)gp;
        asm volatile("global_load_async_to_lds_b128 %0, %1, off"
                     :: "v"(ldsa), "v"(ga) : "memory");
      }
    }
  }

  // ---- per-lane column constants (tiny, L2-resident across all blocks);
  //      these loads overlap the async tile copy ----
  const float4 rw0 = ld4(resp_w + col0);
  const float4 rw1 = ld4(resp_w + col0 + 4);
  const float4 g0  = ld4(gamma_ + col0);
  const float4 g1  = ld4(gamma_ + col0 + 4);
  const float4 be0 = ld4(beta_  + col0);
  const float4 be1 = ld4(beta_  + col0 + 4);

  // full relative-position sum over all offsets except d==0 (w==kW)
  float4 full0 = make_float4(0.f, 0.f, 0.f, 0.f);
  float4 full1 = make_float4(0.f, 0.f, 0.f, 0.f);
#pragma unroll
  for (int w = 0; w < 2 * kW + 1; ++w) {
    if (w == kW) continue;
    full0 = add4(full0, ld4(relpos + w * kD + col0));
    full1 = add4(full1, ld4(relpos + w * kD + col0 + 4));
  }

  if (lane < kRows) respf[lane] = 0.25f * (float)resp[rowbase + lane];

  // prefetch the next tile's embeddings into cache (global_prefetch_b8)
  if (blockIdx.x + 1 < gridDim.x) {
    __builtin_prefetch(emb + (size_t)(rowbase + kRows) * kD + col0, 0, 1);
  }

  // wait for the async tile copy, then make it (and respf) visible
  asm volatile("s_wait_asynccnt 0" ::: "memory");
  __syncthreads();

  // ---- Phase A: fuse biases in place in LDS (each lane RMWs only its own slots) ----
#pragma unroll
  for (int m = 0; m < kRows; ++m) {
    float* tp = tile + m * kRowStride + col0;
    float4 e0 = ld4(tp);
    float4 e1 = ld4(tp + 4);

    const int s = s0 + m;
    float4 b0 = full0, b1 = full1;
    if (s < kW) {                                   // drop offsets hitting pos < 0
      for (int w = 0; w <= kW - 1 - s; ++w) {
        b0 = sub4(b0, ld4(relpos + w * kD + col0));
        b1 = sub4(b1, ld4(relpos + w * kD + col0 + 4));
      }
    }
    if (s > kS - 1 - kW) {                          // drop offsets hitting pos >= S
      for (int w = kS - s + kW; w <= 2 * kW; ++w) {
        b0 = sub4(b0, ld4(relpos + w * kD + col0));
        b1 = sub4(b1, ld4(relpos + w * kD + col0 + 4));
      }
    }
    const float rb = 0.3f * respf[m];
    st4(tp,     biasfuse(e0, rb, rw0, 0.15f, b0));
    st4(tp + 4, biasfuse(e1, rb, rw1, 0.15f, b1));
  }
  __syncthreads();

  // ---- Phase B: partial sums in the exact 16x4 fp32 A-matrix layout, then
  //      one V_WMMA_F32_16X16X4_F32 against an all-ones B reduces all 16 rows.
  //      Lane l<16 owns row l, K=0 (cols 0..63) & K=1 (cols 64..127);
  //      lane l+16 owns row l, K=2 (cols 128..191) & K=3 (cols 192..255).
  const int mrow = lane & 15;
  const int half = lane >> 4;
  const float* trow = tile + mrow * kRowStride + half * 128;

  float ps0 = 0.f, ps1 = 0.f, pq0 = 0.f, pq1 = 0.f;
#pragma unroll
  for (int i = 0; i < 64; i += 4) {
    float4 v = ld4(trow + i);
    ps0 += (v.x + v.y) + (v.z + v.w);
    pq0 += (v.x * v.x + v.y * v.y) + (v.z * v.z + v.w * v.w);
  }
#pragma unroll
  for (int i = 64; i < 128; i += 4) {
    float4 v = ld4(trow + i);
    ps1 += (v.x + v.y) + (v.z + v.w);
    pq1 += (v.x * v.x + v.y * v.y) + (v.z * v.z + v.w * v.w);
  }

  v2f a_sum = {ps0, ps1};
  v2f a_sq  = {pq0, pq1};
  v2f bone  = {1.0f, 1.0f};          // all-ones 4x16 B matrix (layout-independent)
  v8f cz    = {0.f, 0.f, 0.f, 0.f, 0.f, 0.f, 0.f, 0.f};

  // D[m,n] = sum_k A[m,k] -> row sums replicated across all 16 columns
  v8f csum = __builtin_amdgcn_wmma_f32_16x16x4_f32(
      false, a_sum, false, bone, (short)0, cz, false, false);
  v8f csq  = __builtin_amdgcn_wmma_f32_16x16x4_f32(
      false, a_sq,  false, bone, (short)0, cz, false, false);

  // C/D layout: lanes 0-15 hold rows 0..7 in VGPRs 0..7; lanes 16-31 hold rows 8..15.
  if ((lane & 15) == 0) {            // lanes 0 and 16 publish 8 rows each
    const int base = half * 8;
#pragma unroll
    for (int j = 0; j < 8; ++j) {
      const float mean = csum[j] * (1.0f / (float)kD);
      const float ex2  = csq[j]  * (1.0f / (float)kD);
      const float var  = ex2 - mean * mean;
      const float vv   = var + kEps;
      float r = rsqrtf(vv);
      r = r * (1.5f - 0.5f * vv * r * r);   // one Newton step -> full fp32 accuracy
      stats[2 * (base + j) + 0] = mean;
      stats[2 * (base + j) + 1] = r;
    }
  }
  __syncthreads();

  // ---- Phase C: normalize + affine, coalesced b128 stores ----
#pragma unroll
  for (int m = 0; m < kRows; ++m) {
    const float mean = stats[2 * m + 0];
    const float rstd = stats[2 * m + 1];
    const float* tp = tile + m * kRowStride + col0;
    float4 x0 = ld4(tp);
    float4 x1 = ld4(tp + 4);
    float* op = out + (size_t)(rowbase + m) * kD + col0;
    st4(op,     lnorm4(x0, mean, rstd, g0, be0));
    st4(op + 4, lnorm4(x1, mean, rstd, g1, be1));
  }
}

extern "C" void kernel_launch(void* const* d_in, const int* in_sizes, int n_in,
                              void* d_out, int out_size, void* d_ws, size_t ws_size,
                              hipStream_t stream) {
  const float* emb    = (const float*)d_in[0];   // [B,S,D] fp32
  const int*   resp   = (const int*)  d_in[1];   // [B,S]   int32
  const float* relpos = (const float*)d_in[2];   // [2W+1,D] fp32
  const float* rw     = (const float*)d_in[3];   // [D,1]   fp32
  const float* gm     = (const float*)d_in[4];   // [D]
  const float* bt     = (const float*)d_in[5];   // [D]
  float* out = (float*)d_out;

  const int total_rows = in_sizes[1];            // B*S
  const int nblocks = total_rows / kRows;        // 8192 for B=64,S=2048
  rta_fused_kernel<<<nblocks, 32, 0, stream>>>(emb, resp, relpos, rw, gm, bt, out);
}